// WeightedDGeM_37117107372109
// MI455X (gfx1250) — compile-verified
//
#include <hip/hip_runtime.h>
#include <hip/hip_bf16.h>

// mag[b,k,d] = ( (1/w_sum[b,k]) * sum_l a_t[b,k,l] * (|x[b,l,d]|+eps)^p[b,k] )^(1/p[b,k])
// B=4, K=32, L=2048, D=256.  TRANS-bound: 64M exp2 dominate; memory ~9MB total.

#define EPSV 1e-6f

typedef float v2f __attribute__((ext_vector_type(2)));
typedef float v8f __attribute__((ext_vector_type(8)));

#define KQ 32
#define DD 256
#define LL 2048

// v_exp_f32 / v_log_f32 (base-2) via target builtins; header fast-math
// wrappers (__exp2f) are not device-visible on this toolchain.
__device__ __forceinline__ float fast_exp2(float v) { return __builtin_amdgcn_exp2f(v); }
__device__ __forceinline__ float fast_log2(float v) { return __builtin_amdgcn_logf(v); }

// ---------------------------------------------------------------------------
// Kernel 1: per (b, l-chunk) partial sums.
// block = 256 threads (8 waves, wave32), thread d owns one output column.
// LDS holds a[b, 0..31, l0..l0+lc) (shared across all d); p in VGPRs.
// Inner loop per l: 1 v_log_f32 + 32 x (v_mul, v_exp_f32, v_fma).
// partial layout: partial[((b*nchunk + c)*KQ + k)*DD + d]
// ---------------------------------------------------------------------------
__global__ __launch_bounds__(256) void pw_partial_kernel(
    const float* __restrict__ a_t,   // (B,K,L)
    const float* __restrict__ x,     // (B,L,D)
    const float* __restrict__ p,     // (B,K)
    float* __restrict__ partial,
    int nchunk, int lc)
{
    __shared__ float aT[KQ * 128];   // up to lc=128 -> 16 KB
    __shared__ float pS[KQ];

    const int b  = blockIdx.x / nchunk;
    const int c  = blockIdx.x % nchunk;
    const int l0 = c * lc;
    const int d  = threadIdx.x;

    // Stage a-tile into LDS: KQ rows of lc contiguous floats, as float4.
    const int lcv4   = lc >> 2;
    const int total4 = KQ * lcv4;
    for (int i = threadIdx.x; i < total4; i += 256) {
        const int k = i / lcv4;
        const int j = i - k * lcv4;
        ((float4*)aT)[i] =
            *(const float4*)(a_t + ((size_t)b * KQ + k) * LL + l0 + 4 * j);
    }
    if (threadIdx.x < KQ) pS[threadIdx.x] = fmaxf(p[b * KQ + threadIdx.x], 1.0f);
    __syncthreads();

    float pk[KQ], acc[KQ];
#pragma unroll
    for (int k = 0; k < KQ; ++k) { pk[k] = pS[k]; acc[k] = 0.0f; }

    const float* xp = x + ((size_t)b * LL + l0) * DD + d;

    for (int ll = 0; ll < lc; ++ll) {
        // gfx1250 prefetch path (global_prefetch_b8) on the streaming x reads
        __builtin_prefetch(xp + (ll + 8) * DD, 0, 1);
        const float t = fast_log2(fabsf(xp[(size_t)ll * DD]) + EPSV);
#pragma unroll
        for (int k = 0; k < KQ; ++k) {
            acc[k] += aT[k * lc + ll] * fast_exp2(pk[k] * t);
        }
    }

    float* po = partial + (((size_t)b * nchunk + c) * KQ) * DD + d;
#pragma unroll
    for (int k = 0; k < KQ; ++k) po[(size_t)k * DD] = acc[k];
}

// ---------------------------------------------------------------------------
// Kernel 2: reduce over chunks with V_WMMA_F32_16X16X4_F32 (full f32).
// D = A(ones,16x4) x B(4 chunks x 16 outputs) + C  accumulated over g.
// Each wave produces 16 contiguous outputs n = wave*16 + lane%16
// (n -> b = n/(K*D), k = (n/D)%K, d = n%D; 16 | D so b,k uniform per wave).
// B-operand f32 4x16 layout: VGPR0 = rows {2*half}, VGPR1 = rows {2*half+1},
// half = lane/16. C/D rows are all equal (A rows identical), and the C/D
// layout (M=0 lanes 0-15, M=8 lanes 16-31 in VGPR0) makes c[0] = colsum(n)
// for every lane. Finalize: (num/w_sum)^(1/p) via exp2/log2.
// ---------------------------------------------------------------------------
__global__ __launch_bounds__(256) void pw_reduce_kernel(
    const float* __restrict__ partial,
    const float* __restrict__ w_sum,  // (B,K)
    const float* __restrict__ p,      // (B,K)
    float* __restrict__ out,          // (B,K,D)
    int nchunk)
{
    const int lane = threadIdx.x & 31;
    const int half = lane >> 4;
    const int wave = (blockIdx.x * blockDim.x + threadIdx.x) >> 5;

    const int n = wave * 16 + (lane & 15);
    const int d = n % DD;
    const int k = (n / DD) % KQ;
    const int b = n / (DD * KQ);

    const float*  pb      = partial + ((size_t)b * nchunk) * KQ * DD
                                    + (size_t)k * DD + d;
    const size_t  cstride = (size_t)KQ * DD;

    v2f ones; ones.x = 1.0f; ones.y = 1.0f;
    v8f c = {};

    for (int g = 0; g < nchunk; g += 4) {
        v2f bv;
        bv.x = pb[(size_t)(g + 2 * half + 0) * cstride];
        bv.y = pb[(size_t)(g + 2 * half + 1) * cstride];
        // D = ones(16x4) x B(4x16) + C   -> column sums, accumulated in C
        c = __builtin_amdgcn_wmma_f32_16x16x4_f32(
                /*neg_a=*/false, ones, /*neg_b=*/false, bv,
                /*c_mod=*/(short)0, c, /*reuse_a=*/false, /*reuse_b=*/false);
    }

    const float num = c[0];   // valid in every lane for column n = lane%16
    if (half == 0) {
        const float pkv   = fmaxf(p[b * KQ + k], 1.0f);
        const float ws    = w_sum[b * KQ + k];
        const float ratio = num / ws;
        out[n] = fast_exp2(fast_log2(ratio) * (1.0f / pkv));
    }
}

// ---------------------------------------------------------------------------
extern "C" void kernel_launch(void* const* d_in, const int* in_sizes, int n_in,
                              void* d_out, int out_size, void* d_ws, size_t ws_size,
                              hipStream_t stream) {
    const float* a_t  = (const float*)d_in[0];  // (4,32,2048)
    const float* x    = (const float*)d_in[1];  // (4,2048,256)
    const float* wsum = (const float*)d_in[2];  // (4,32,1)
    const float* p    = (const float*)d_in[3];  // (4,32,1)
    float* out        = (float*)d_out;          // (4,32,256)
    float* partial    = (float*)d_ws;

    constexpr int B = 4;

    // Prefer 64 chunks (8 MB partials, 256 WGs = 2048 waves); shrink if ws small.
    int nchunk = 64;
    while ((size_t)B * nchunk * KQ * DD * sizeof(float) > ws_size && nchunk > 16)
        nchunk >>= 1;
    const int lc = LL / nchunk;   // 32..128

    dim3 g1(B * nchunk), blk(256);
    pw_partial_kernel<<<g1, blk, 0, stream>>>(a_t, x, p, partial, nchunk, lc);

    const int total = B * KQ * DD;              // 32768 outputs
    dim3 g2((total / 16 * 32) / 256);           // 16 outputs per wave
    pw_reduce_kernel<<<g2, blk, 0, stream>>>(partial, wsum, p, out, nchunk);
}